// GCNConv_687194767734
// MI455X (gfx1250) — compile-verified
//
#include <hip/hip_runtime.h>

typedef __attribute__((ext_vector_type(2))) float v2f;
typedef __attribute__((ext_vector_type(8))) float v8f;

#define C_DIM   128
#define MAX_DEG 16
#define WPAIR_STRIDE 144   // float2 per K-pair row: 288 dwords = 32 mod 64 banks
                           // -> hi half-wave (kk+2) hits banks 32..63 vs 0..31: conflict-free

// ---------------------------------------------------------------------------
// Kernel 1: h[i,:] = (x[i,:] @ W) * rsqrt(deg_i + 1)
// 8 waves / block, each wave computes a 16-row x 128-col tile via
// V_WMMA_F32_16X16X4_F32 (full fp32, matches reference dtype).
// W is staged in LDS K-pair interleaved: LDS float2 {W[2p][n], W[2p+1][n]}
// at pair-row p, col n  ->  every B fragment is ONE ds_load_b64 straight into
// an even-aligned VGPR pair (no repack movs feeding the WMMA).
// ---------------------------------------------------------------------------
__global__ __launch_bounds__(256) void gcn_gemm(const float* __restrict__ x,
                                                const int*   __restrict__ edge,
                                                const float* __restrict__ W,
                                                float*       __restrict__ h,
                                                int nStrips) {
    __shared__ float Wlds[64 * WPAIR_STRIDE * 2];   // 72 KB (pad region never read)

    // Cooperative interleaved stage: 64 pair-rows x 64 float4 = 4096 float4.
    // Thread q handles pair-row p = q>>6, col-pair c = q&63 (n = 2c):
    // reads W[2p][n..n+1], W[2p+1][n..n+1], writes {a.x, b.x, a.y, b.y}.
    {
        const int tid = threadIdx.x;
#pragma unroll
        for (int i = 0; i < 16; ++i) {
            const int q = tid + 256 * i;            // 0..4095
            const int p = q >> 6;
            const int c = q & 63;
            const float2 ga = *(const float2*)(W + (size_t)(2 * p)     * C_DIM + 2 * c);
            const float2 gb = *(const float2*)(W + (size_t)(2 * p + 1) * C_DIM + 2 * c);
            float4 v; v.x = ga.x; v.y = gb.x; v.z = ga.y; v.w = gb.y;
            ((float4*)Wlds)[p * (WPAIR_STRIDE / 2) + c] = v;
        }
    }
    __syncthreads();

    const int wave  = threadIdx.x >> 5;
    const int lane  = threadIdx.x & 31;
    const int strip = blockIdx.x * 8 + wave;        // one 16-row strip per wave
    if (strip >= nStrips) return;                   // whole-wave exit: EXEC stays all-1s

    const int m  = lane & 15;                       // row within strip (A) / col within tile (B)
    const int hi = lane >> 4;                       // K-pair select per ISA layout
    const int rowBase = strip * 16;

    const float* xrow = x + (size_t)(rowBase + m) * C_DIM;
    // Per-lane B base: pair-row (hi), col m. Advances by 2 pair-rows per k0 step.
    const float* wb0 = Wlds + ((size_t)hi * WPAIR_STRIDE + m) * 2;

    v8f c[8] = {};                                  // 8 accumulators: 16x128 output

    for (int k0 = 0; k0 < C_DIM; k0 += 4) {
        const int kk = k0 + 2 * hi;
        // A 16x4 f32: lane(m,hi) holds A[m][kk], A[m][kk+1]  (kk even -> 8B aligned)
        v2f a = *(const v2f*)(xrow + kk);
        const float* wb = wb0 + (size_t)(k0 >> 1) * WPAIR_STRIDE * 2;
#pragma unroll
        for (int t = 0; t < 8; ++t) {
            // B 4x16 f32: one ds_load_b64 = {W[kk][n0+m], W[kk+1][n0+m]}
            v2f b = *(const v2f*)(wb + t * 32);
            c[t] = __builtin_amdgcn_wmma_f32_16x16x4_f32(
                false, a, false, b, (short)0, c[t], false, false);
        }
    }

    // inv_sqrt_deg for this wave's 16 rows: lane (and lane+16) counts row rowBase+m.
    const int* e = edge + (size_t)(rowBase + m) * MAX_DEG;
    int cnt = 0;
#pragma unroll
    for (int k = 0; k < MAX_DEG; ++k) cnt += (e[k] >= 0) ? 1 : 0;
    const float sv = rsqrtf((float)cnt + 1.0f);

    // C/D layout: lanes 0-15 VGPR r -> row r; lanes 16-31 VGPR r -> row r+8.
#pragma unroll
    for (int r = 0; r < 8; ++r) {
        const float sc  = __shfl(sv, r + 8 * hi, 32);     // scale for this lane's row
        const int   row = rowBase + r + 8 * hi;
        float* hrow = h + (size_t)row * C_DIM;
#pragma unroll
        for (int t = 0; t < 8; ++t)
            hrow[t * 16 + m] = c[t][r] * sc;              // 64B-coalesced across lanes
    }
}

// ---------------------------------------------------------------------------
// Kernel 2: out[i,:] = (sum_{j in nbrs(i)} h[j,:] + h[i,:]) * rsqrt(deg_i + 1)
// One wave per node: each lane owns 4 channels (float4); every neighbor-row
// gather is a fully coalesced 512B read served from global L2 (h is 51MB vs
// 192MB L2). Next-row global_prefetch_b8 hides gather latency.
// ---------------------------------------------------------------------------
__global__ __launch_bounds__(256) void gcn_agg(const float* __restrict__ h,
                                               const int*   __restrict__ edge,
                                               float*       __restrict__ out,
                                               int nNodes) {
    const int lane = threadIdx.x & 31;
    const int node = blockIdx.x * 8 + (threadIdx.x >> 5);
    if (node >= nNodes) return;

    int idx = -1;
    if (lane < MAX_DEG) idx = edge[(size_t)node * MAX_DEG + lane];

    const unsigned long long valid = __ballot(idx >= 0);
    const float s = rsqrtf((float)__popcll(valid) + 1.0f);

    const float4* h4 = (const float4*)h;
    float4 acc = h4[(size_t)node * 32 + lane];            // self contribution

    int jnext = __shfl(idx, 0, 32);
#pragma unroll 4
    for (int k = 0; k < MAX_DEG; ++k) {
        const int j = jnext;
        jnext = (k + 1 < MAX_DEG) ? __shfl(idx, k + 1, 32) : -1;
        if (jnext >= 0)
            __builtin_prefetch(&h4[(size_t)jnext * 32 + lane], 0, 3);  // global_prefetch_b8
        if (j >= 0) {                                     // uniform branch (broadcast j)
            const float4 v = h4[(size_t)j * 32 + lane];
            acc.x += v.x; acc.y += v.y; acc.z += v.z; acc.w += v.w;
        }
    }

    float4 o;
    o.x = acc.x * s; o.y = acc.y * s; o.z = acc.z * s; o.w = acc.w * s;
    ((float4*)out)[(size_t)node * 32 + lane] = o;
}

// ---------------------------------------------------------------------------
extern "C" void kernel_launch(void* const* d_in, const int* in_sizes, int n_in,
                              void* d_out, int out_size, void* d_ws, size_t ws_size,
                              hipStream_t stream) {
    const float* x    = (const float*)d_in[0];   // [N, 128] f32
    const int*   edge = (const int*)  d_in[1];   // [N, 16]  i32
    const float* W    = (const float*)d_in[2];   // [128,128] f32
    float*       out  = (float*)d_out;           // [N, 128] f32
    float*       h    = (float*)d_ws;            // scratch: N*128 f32 (51.2 MB)

    const int N       = in_sizes[0] / C_DIM;     // 100000
    const int nStrips = (N + 15) / 16;           // 6250 (exact: EXEC all-1s per wave)

    const int gemmBlocks = (nStrips + 7) / 8;
    gcn_gemm<<<gemmBlocks, 256, 0, stream>>>(x, edge, W, h, nStrips);

    const int aggBlocks = (N + 7) / 8;
    gcn_agg<<<aggBlocks, 256, 0, stream>>>(h, edge, out, N);
}